// Wasserstein_Quantizer_17995912970294
// MI455X (gfx1250) — compile-verified
//
#include <hip/hip_runtime.h>
#include <math.h>

#define DIM   32
#define N_VEC 8192
#define K_CB  16384
#define Q_SZ  65536

typedef __attribute__((ext_vector_type(2))) float v2f;
typedef __attribute__((ext_vector_type(8))) float v8f;

// ---- workspace layout (float units) ----
#define WS_ZF     0        // 262144
#define WS_EMBN   262144   // 16384
#define WS_TOKEN  278528   // 8192 (int)
#define WS_HIST   286720   // 16384 (uint)
#define WS_SUMSQ  303104   // 1
#define WS_UTIL   303105   // 1
#define WS_PERP   303106   // 1
#define WS_GC     303136   // 1024
#define WS_SC     304160   // 32
#define WS_GZ     304192   // 1024
#define WS_SZ     305216   // 32
#define WS_TOTAL  305248

// ---------------- init: zero all atomic-accumulation regions ----------------
__global__ void wq_init_kernel(float* ws) {
    int t = blockIdx.x * 256 + threadIdx.x;
    if (t < (WS_TOTAL - WS_HIST)) ws[WS_HIST + t] = 0.0f;
}

// ---------------- z [B,C,H,W] -> zf [N, 32] ----------------
__global__ void wq_transpose_kernel(const float* __restrict__ z, float* __restrict__ zf) {
    int t = blockIdx.x * 256 + threadIdx.x;   // 0..262143
    int n = t >> 5, c = t & 31;
    int b = n >> 10, p = n & 1023;            // H*W = 1024
    zf[t] = z[b * 32768 + c * 1024 + p];
}

// ---------------- ||emb_k||^2 ----------------
__global__ void wq_embnorm_kernel(const float* __restrict__ emb, float* __restrict__ en) {
    int k = blockIdx.x * 256 + threadIdx.x;
    if (k < K_CB) {
        const float4* p = (const float4*)(emb + (size_t)k * DIM);
        float s = 0.0f;
#pragma unroll
        for (int q = 0; q < 8; ++q) { float4 v = p[q]; s += v.x*v.x + v.y*v.y + v.z*v.z + v.w*v.w; }
        en[k] = s;
    }
}

// ---------------- WMMA distance + argmin ----------------
// block = 256 (8 waves); block owns 32 rows (two 16-row A fragments per wave);
// wave w scans codes [w*2048,(w+1)*2048), 16 codes per chunk, B prefetched one
// chunk ahead so WMMAs never stall on loads.
__global__ __launch_bounds__(256) void wq_argmin_kernel(const float* __restrict__ zf,
                                                        const float* __restrict__ emb,
                                                        const float* __restrict__ embn,
                                                        int* __restrict__ token) {
    __shared__ float sd[8][32];
    __shared__ int   sc[8][32];
    const int rowbase = blockIdx.x * 32;
    const int wave = threadIdx.x >> 5;
    const int lane = threadIdx.x & 31;
    const int half = lane >> 4;        // K-pair select
    const int lrow = lane & 15;        // M for A-frag, N for B-frag

    // Two A fragments (rows rowbase..+15 and rowbase+16..+31), loaded once.
    const v2f* arow0 = (const v2f*)(zf + (size_t)(rowbase + lrow) * DIM);
    const v2f* arow1 = (const v2f*)(zf + (size_t)(rowbase + 16 + lrow) * DIM);
    v2f a0[8], a1[8];
#pragma unroll
    for (int ks = 0; ks < 8; ++ks) { a0[ks] = arow0[ks * 2 + half]; a1[ks] = arow1[ks * 2 + half]; }

    float bestd0[8], bestd1[8];
    int   bestc0[8], bestc1[8];
#pragma unroll
    for (int v = 0; v < 8; ++v) {
        bestd0[v] = 3.0e38f; bestc0[v] = 0;
        bestd1[v] = 3.0e38f; bestc1[v] = 0;
    }

    const int col0 = wave * 2048;

    // prologue: load B fragments + norm for chunk 0
    v2f b[8];
    {
        const v2f* brow = (const v2f*)(emb + (size_t)(col0 + lrow) * DIM);
#pragma unroll
        for (int ks = 0; ks < 8; ++ks) b[ks] = brow[ks * 2 + half];
    }
    float en = embn[col0 + lrow];

    for (int chunk = 0; chunk < 128; ++chunk) {
        const int col  = col0 + chunk * 16 + lrow;
        const int ncol = col0 + ((chunk + 1) & 127) * 16 + lrow;  // wraps (dummy last prefetch)

        // prefetch next chunk's B fragments + norm (independent regs -> pipelined)
        v2f bn[8];
        const v2f* nrow = (const v2f*)(emb + (size_t)ncol * DIM);
#pragma unroll
        for (int ks = 0; ks < 8; ++ks) bn[ks] = nrow[ks * 2 + half];
        float enn = embn[ncol];

        v8f acc0 = {0.f,0.f,0.f,0.f,0.f,0.f,0.f,0.f};
        v8f acc1 = {0.f,0.f,0.f,0.f,0.f,0.f,0.f,0.f};
#pragma unroll
        for (int ks = 0; ks < 8; ++ks) {
            acc0 = __builtin_amdgcn_wmma_f32_16x16x4_f32(false, a0[ks], false, b[ks],
                                                         (short)0, acc0, false, false);
            acc1 = __builtin_amdgcn_wmma_f32_16x16x4_f32(false, a1[ks], false, b[ks],
                                                         (short)0, acc1, false, false);
        }
#pragma unroll
        for (int v = 0; v < 8; ++v) {
            float d2 = fmaf(-2.0f, acc0[v], en);    // + ||zf||^2 row-constant: argmin-invariant
            if (d2 < bestd0[v]) { bestd0[v] = d2; bestc0[v] = col; }
            float d3 = fmaf(-2.0f, acc1[v], en);
            if (d3 < bestd1[v]) { bestd1[v] = d3; bestc1[v] = col; }
        }
#pragma unroll
        for (int ks = 0; ks < 8; ++ks) b[ks] = bn[ks];
        en = enn;
    }

    // reduce across the 16 lanes of each half (xor masks <16 stay in-half)
#pragma unroll
    for (int v = 0; v < 8; ++v) {
#pragma unroll
        for (int m = 8; m >= 1; m >>= 1) {
            float od = __shfl_xor(bestd0[v], m, 32);
            int   oc = __shfl_xor(bestc0[v], m, 32);
            if (od < bestd0[v] || (od == bestd0[v] && oc < bestc0[v])) { bestd0[v] = od; bestc0[v] = oc; }
            float pd = __shfl_xor(bestd1[v], m, 32);
            int   pc = __shfl_xor(bestc1[v], m, 32);
            if (pd < bestd1[v] || (pd == bestd1[v] && pc < bestc1[v])) { bestd1[v] = pd; bestc1[v] = pc; }
        }
    }
    if (lrow == 0) {
#pragma unroll
        for (int v = 0; v < 8; ++v) {
            sd[wave][v + 8 * half]      = bestd0[v];
            sc[wave][v + 8 * half]      = bestc0[v];
            sd[wave][16 + v + 8 * half] = bestd1[v];
            sc[wave][16 + v + 8 * half] = bestc1[v];
        }
    }
    __syncthreads();
    if (threadIdx.x < 32) {
        float bd = sd[0][threadIdx.x]; int bc = sc[0][threadIdx.x];
        for (int w = 1; w < 8; ++w) {
            float od = sd[w][threadIdx.x]; int oc = sc[w][threadIdx.x];
            if (od < bd || (od == bd && oc < bc)) { bd = od; bc = oc; }
        }
        token[rowbase + threadIdx.x] = bc;
    }
}

// ---------------- gather z_q, write output (b,c,h,w), accumulate sum of squared diffs ----------------
__global__ __launch_bounds__(256) void wq_gather_kernel(const float* __restrict__ zf,
                                                        const float* __restrict__ emb,
                                                        const int* __restrict__ token,
                                                        float* __restrict__ out,
                                                        float* __restrict__ sumsq) {
    __shared__ float red[256];
    int t = blockIdx.x * 256 + threadIdx.x;   // 0..262143
    int n = t >> 5, c = t & 31;
    float q = emb[(size_t)token[n] * DIM + c];
    float d = q - zf[t];
    int b = n >> 10, p = n & 1023;
    out[b * 32768 + c * 1024 + p] = q;        // straight-through forward == z_q
    red[threadIdx.x] = d * d;
    __syncthreads();
    for (int s = 128; s > 0; s >>= 1) {
        if (threadIdx.x < s) red[threadIdx.x] += red[threadIdx.x + s];
        __syncthreads();
    }
    if (threadIdx.x == 0) atomicAdd(sumsq, red[0]);
}

// ---------------- token histogram ----------------
__global__ void wq_hist_kernel(const int* __restrict__ token, unsigned* __restrict__ hist) {
    int n = blockIdx.x * 256 + threadIdx.x;
    if (n < N_VEC) atomicAdd(&hist[token[n]], 1u);
}

// ---------------- utilization + perplexity ----------------
__global__ void wq_stats_kernel(const unsigned* __restrict__ hist, float* util, float* perp) {
    __shared__ float scnt[256], sent[256];
    float cnt = 0.0f, ent = 0.0f;
    for (int k = threadIdx.x; k < K_CB; k += 256) {
        unsigned h = hist[k];
        if (h > 0u) {
            cnt += 1.0f;
            float p = (float)h / (float)N_VEC;
            ent -= p * logf(p + 1e-10f);
        }
    }
    scnt[threadIdx.x] = cnt; sent[threadIdx.x] = ent;
    __syncthreads();
    for (int s = 128; s > 0; s >>= 1) {
        if (threadIdx.x < s) { scnt[threadIdx.x] += scnt[threadIdx.x + s]; sent[threadIdx.x] += sent[threadIdx.x + s]; }
        __syncthreads();
    }
    if (threadIdx.x == 0) { *util = scnt[0] / (float)K_CB; *perp = expf(sent[0]); }
}

// ---------------- Gram matrix + column sums: rows < nA from A else from B ----------------
__global__ __launch_bounds__(1024) void wq_gram_kernel(const float* __restrict__ A, int nA,
                                                       const float* __restrict__ B,
                                                       int rowsPerBlock,
                                                       float* __restrict__ G, float* __restrict__ S) {
    __shared__ float tile[128 * 32];
    const int i = threadIdx.x >> 5, j = threadIdx.x & 31;
    const int row0 = blockIdx.x * rowsPerBlock;
    float acc = 0.0f, rsum = 0.0f;
    for (int t0 = 0; t0 < rowsPerBlock; t0 += 128) {
        for (int e = threadIdx.x; e < 128 * 32; e += 1024) {
            int r = row0 + t0 + (e >> 5);
            int k = e & 31;
            tile[e] = (r < nA) ? A[(size_t)r * DIM + k] : B[(size_t)r * DIM + k];
        }
        __syncthreads();
#pragma unroll 4
        for (int r = 0; r < 128; ++r) {
            float xj = tile[r * 32 + j];
            acc += tile[r * 32 + i] * xj;
            if (i == 0) rsum += xj;
        }
        __syncthreads();
    }
    atomicAdd(&G[i * 32 + j], acc);
    if (i == 0) atomicAdd(&S[j], rsum);
}

// ---------------- Newton-Schulz sqrtm of 32x32 SPD matrix in LDS ----------------
// result: sqrt(A) ~= Y * sqrt(c0), c0 left in *s_c0
__device__ __forceinline__ void wq_ns_sqrt(const float* A, float* Y, float* Z, float* T,
                                           int i, int j, float* s_c0) {
    const int ij = i * 32 + j;
    if (threadIdx.x == 0) {
        float tr = 0.0f;
        for (int k = 0; k < 32; ++k) tr += A[k * 32 + k];
        *s_c0 = tr;
    }
    __syncthreads();
    const float inv_c0 = 1.0f / (*s_c0);
    Y[ij] = A[ij] * inv_c0;
    Z[ij] = (i == j) ? 1.0f : 0.0f;
    __syncthreads();
    for (int it = 0; it < 30; ++it) {
        float t = (i == j) ? 3.0f : 0.0f;
        for (int k = 0; k < 32; ++k) t -= Z[i * 32 + k] * Y[k * 32 + j];
        T[ij] = t;
        __syncthreads();
        float ny = 0.0f, nz = 0.0f;
        for (int k = 0; k < 32; ++k) {
            ny += Y[i * 32 + k] * T[k * 32 + j];
            nz += T[i * 32 + k] * Z[k * 32 + j];
        }
        __syncthreads();
        Y[ij] = 0.5f * ny;
        Z[ij] = 0.5f * nz;
        __syncthreads();
    }
}

// ---------------- finalize: wloss + all scalar outputs ----------------
__global__ __launch_bounds__(1024) void wq_finalize_kernel(const float* __restrict__ Gc, const float* __restrict__ Sc,
                                                           const float* __restrict__ Gz, const float* __restrict__ Sz,
                                                           const float* __restrict__ sumsq,
                                                           const float* __restrict__ util, const float* __restrict__ perp,
                                                           float* __restrict__ out_scalars) {
    __shared__ float A[1024], Cc[1024], Y[1024], Z[1024], T[1024], Sm[1024];
    __shared__ float diag[32];
    __shared__ float sh[8];
    const int i = threadIdx.x >> 5, j = threadIdx.x & 31, ij = threadIdx.x;
    const float nC = (float)K_CB, nZ = (float)Q_SZ;

    float mucj = Sc[j] / nC, muci = Sc[i] / nC;
    float muzj = Sz[j] / nZ, muzi = Sz[i] / nZ;
    float covc = (Gc[ij] - nC * muci * mucj) / (nC - 1.0f);
    float covz = (Gz[ij] - nZ * muzi * muzj) / (nZ - 1.0f);

    // std = max_j sqrt(var_j(embedding)) (unbiased), pre-scaling
    if (i == j) diag[i] = sqrtf(fmaxf(covc, 0.0f));
    __syncthreads();
    if (threadIdx.x == 0) {
        float m = 0.0f;
        for (int k = 0; k < 32; ++k) m = fmaxf(m, diag[k]);
        sh[0] = m;
    }
    __syncthreads();
    const float inv = 1.0f / (sh[0] + 1e-8f);
    const float inv2 = inv * inv;
    A[ij]  = covz * inv2 + ((i == j) ? 1e-6f : 0.0f);   // scaled z_cov + eps*I
    Cc[ij] = covc * inv2 + ((i == j) ? 1e-6f : 0.0f);   // scaled c_cov + eps*I
    __syncthreads();
    if (i == 0) diag[j] = (muzj - mucj) * inv;          // scaled mean diff (reuse)
    __syncthreads();
    if (threadIdx.x == 0) {
        float pm = 0.0f, tr = 0.0f;
        for (int k = 0; k < 32; ++k) { pm += diag[k] * diag[k]; tr += A[k * 32 + k] + Cc[k * 32 + k]; }
        sh[1] = pm;   // part_mean
        sh[2] = tr;   // trace(z_cov) + trace(c_cov)
    }
    __syncthreads();

    // S = sqrtm(z_cov)
    wq_ns_sqrt(A, Y, Z, T, i, j, &sh[3]);
    const float sc1 = sqrtf(sh[3]);
    Sm[ij] = Y[ij] * sc1;
    __syncthreads();
    // covariance = S @ c_cov @ S
    float t1 = 0.0f;
    for (int k = 0; k < 32; ++k) t1 += Sm[i * 32 + k] * Cc[k * 32 + j];
    __syncthreads();
    Y[ij] = t1;
    __syncthreads();
    float t2 = 0.0f;
    for (int k = 0; k < 32; ++k) t2 += Y[i * 32 + k] * Sm[k * 32 + j];
    __syncthreads();
    A[ij] = t2;
    __syncthreads();
    // trace(sqrtm(covariance)) == sum(sqrt(eig))
    wq_ns_sqrt(A, Y, Z, T, i, j, &sh[4]);
    if (threadIdx.x == 0) {
        float trY = 0.0f;
        for (int k = 0; k < 32; ++k) trY += Y[k * 32 + k];
        float tr_sqrt = sqrtf(sh[4]) * trY;
        float part_cov = fmaxf(sh[2] - 2.0f * tr_sqrt, 0.0f);
        float wloss = sqrtf(sh[1] + part_cov + 1e-10f);
        float ss = sumsq[0];
        float mse = ss / 262144.0f;                 // N_VEC * DIM
        out_scalars[0] = 1.25f * mse + wloss;       // (BETA+ALPHA)*mse + GAMMA*wloss
        out_scalars[1] = wloss;
        out_scalars[2] = ss / (float)N_VEC;         // quant_error
        out_scalars[3] = util[0];
        out_scalars[4] = perp[0];
    }
}

extern "C" void kernel_launch(void* const* d_in, const int* in_sizes, int n_in,
                              void* d_out, int out_size, void* d_ws, size_t ws_size,
                              hipStream_t stream) {
    const float* z     = (const float*)d_in[0];   // [8,32,32,32]
    const float* emb   = (const float*)d_in[1];   // [16384,32]
    const float* queue = (const float*)d_in[2];   // [65536,32]
    float* out = (float*)d_out;                   // 262144 z_q_out + 5 scalars
    float* ws  = (float*)d_ws;

    float*    zf    = ws + WS_ZF;
    float*    en    = ws + WS_EMBN;
    int*      token = (int*)(ws + WS_TOKEN);
    unsigned* hist  = (unsigned*)(ws + WS_HIST);
    float*    sumsq = ws + WS_SUMSQ;

    wq_init_kernel<<<73, 256, 0, stream>>>(ws);
    wq_transpose_kernel<<<1024, 256, 0, stream>>>(z, zf);
    wq_embnorm_kernel<<<64, 256, 0, stream>>>(emb, en);
    wq_argmin_kernel<<<256, 256, 0, stream>>>(zf, emb, en, token);
    wq_gather_kernel<<<1024, 256, 0, stream>>>(zf, emb, token, out, sumsq);
    wq_hist_kernel<<<32, 256, 0, stream>>>(token, hist);
    wq_stats_kernel<<<1, 256, 0, stream>>>(hist, ws + WS_UTIL, ws + WS_PERP);
    wq_gram_kernel<<<16, 1024, 0, stream>>>(emb, K_CB, emb, 1024, ws + WS_GC, ws + WS_SC);
    wq_gram_kernel<<<64, 1024, 0, stream>>>(zf, N_VEC, queue, 1024, ws + WS_GZ, ws + WS_SZ);
    wq_finalize_kernel<<<1, 1024, 0, stream>>>(ws + WS_GC, ws + WS_SC, ws + WS_GZ, ws + WS_SZ,
                                               sumsq, ws + WS_UTIL, ws + WS_PERP, out + 262144);
}